// AttnDecoderRNN_70342974374383
// MI455X (gfx1250) — compile-verified
//
#include <hip/hip_runtime.h>
#include <hip/hip_bf16.h>
#include <math.h>

// Problem constants (match reference)
#define H_ 1024
#define F_ 2048
#define V_ 50257

typedef float v2f __attribute__((ext_vector_type(2)));
typedef float v8f __attribute__((ext_vector_type(8)));

// ---------------------------------------------------------------------------
// Kernel 1: embedding gather, build cat1=[emb,h] and init cat2=[emb, 0]
// ---------------------------------------------------------------------------
__global__ void prep_kernel(const int* __restrict__ tok,
                            const float* __restrict__ emb,
                            const float* __restrict__ hid,
                            float* __restrict__ cat1,
                            float* __restrict__ cat2) {
  int i = blockIdx.x * blockDim.x + threadIdx.x;   // 0 .. 2H-1
  int t = tok[0];
  if (i < H_) {
    float e = emb[(size_t)t * H_ + i];
    cat1[i] = e;
    cat2[i] = e;
    cat2[H_ + i] = 0.0f;     // zero accumulation target for attn_applied
  } else if (i < 2 * H_) {
    cat1[i] = hid[i - H_];
  }
}

// ---------------------------------------------------------------------------
// Generic wave-per-row GEMV: out[r] = act(dot(W[r,:], v) + b[r])
// ncols must be a multiple of 128 (float4 x 32 lanes). act: 0=none, 1=relu
// ---------------------------------------------------------------------------
__global__ void gemv_f32(const float* __restrict__ W,
                         const float* __restrict__ v,
                         const float* __restrict__ b,
                         float* __restrict__ out,
                         int nrows, int ncols, int act) {
  int wave = (int)((blockIdx.x * blockDim.x + threadIdx.x) >> 5);
  int lane = threadIdx.x & 31;
  if (wave >= nrows) return;
  const float4* Wr = (const float4*)(W + (size_t)wave * ncols);
  const float4* vv = (const float4*)v;
  int nc4 = ncols >> 2;
  float s = 0.0f;
  for (int j = lane; j < nc4; j += 32) {
    float4 w = Wr[j];
    float4 x = vv[j];
    s += w.x * x.x + w.y * x.y + w.z * x.z + w.w * x.w;
  }
#pragma unroll
  for (int off = 16; off > 0; off >>= 1) s += __shfl_down(s, off, 32);
  if (lane == 0) {
    s += b[wave];
    if (act) s = fmaxf(s, 0.0f);
    out[wave] = s;
  }
}

// ---------------------------------------------------------------------------
// Kernel 3: softmax over n (single block, 256 threads)
// ---------------------------------------------------------------------------
__global__ void softmax_k(const float* __restrict__ x, float* __restrict__ y,
                          int n) {
  __shared__ float red[256];
  int tid = threadIdx.x;
  float m = -3.4e38f;
  for (int i = tid; i < n; i += 256) m = fmaxf(m, x[i]);
  red[tid] = m; __syncthreads();
  for (int s = 128; s > 0; s >>= 1) {
    if (tid < s) red[tid] = fmaxf(red[tid], red[tid + s]);
    __syncthreads();
  }
  float mx = red[0]; __syncthreads();
  float sum = 0.0f;
  for (int i = tid; i < n; i += 256) {
    float e = expf(x[i] - mx);
    y[i] = e;
    sum += e;
  }
  red[tid] = sum; __syncthreads();
  for (int s = 128; s > 0; s >>= 1) {
    if (tid < s) red[tid] += red[tid + s];
    __syncthreads();
  }
  float inv = 1.0f / red[0];
  for (int i = tid; i < n; i += 256) y[i] *= inv;
}

// ---------------------------------------------------------------------------
// Kernel 4: attn_applied[h] += sum_f w[f]*enc[f,h]  (F split over blockIdx.y)
// ---------------------------------------------------------------------------
__global__ void attn_apply(const float* __restrict__ enc,
                           const float* __restrict__ w,
                           float* __restrict__ acc) {
  int h = blockIdx.x * blockDim.x + threadIdx.x;     // 0..H-1 (gridDim.x = 4)
  int f0 = blockIdx.y * (F_ / 8);
  float s = 0.0f;
  for (int f = f0; f < f0 + F_ / 8; ++f) s += w[f] * enc[(size_t)f * H_ + h];
  atomicAdd(&acc[h], s);
}

// ---------------------------------------------------------------------------
// Kernel 7: GRU gate fusion (torch gate order r,z,n)
// ---------------------------------------------------------------------------
__global__ void gru_gates(const float* __restrict__ gx,
                          const float* __restrict__ gh,
                          const float* __restrict__ hid,
                          float* __restrict__ hn_out,
                          float* __restrict__ hn_ws) {
  int i = blockIdx.x * blockDim.x + threadIdx.x;
  if (i >= H_) return;
  float r = 1.0f / (1.0f + expf(-(gx[i] + gh[i])));
  float z = 1.0f / (1.0f + expf(-(gx[H_ + i] + gh[H_ + i])));
  float n = tanhf(gx[2 * H_ + i] + r * gh[2 * H_ + i]);
  float hv = (1.0f - z) * n + z * hid[i];
  hn_out[i] = hv;
  hn_ws[i] = hv;   // 16B-aligned copy for vectorized reads in outproj
}

// ---------------------------------------------------------------------------
// Kernel 8: output projection logits[v] = dot(out_W[v,:], h) + out_b[v]
// 16 rows per wave via V_WMMA_F32_16X16X4_F32.
//   A (16x4 f32): lanes 0-15 hold {K0,K1}, lanes 16-31 hold {K2,K3}  (ISA 7.12.2)
//   B (4x16 f32): h broadcast across all 16 columns -> every column of D equal
// K-order permuted identically in A and B (dot product order-invariant):
//   logical {K0,K1,K2,K3} <- memory {kb,kb+1,kb+4,kb+5} (1st wmma)
//                            memory {kb+2,kb+3,kb+6,kb+7} (2nd wmma)
// so each lane issues one coalesced-per-row float4 load per 8 K values.
// ---------------------------------------------------------------------------
__global__ void outproj_wmma(const float* __restrict__ W,
                             const float* __restrict__ h,
                             const float* __restrict__ bias,
                             float* __restrict__ out) {
  const int lane = threadIdx.x & 31;
  const int tile = (int)((blockIdx.x * blockDim.x + threadIdx.x) >> 5);
  if (tile * 16 >= V_) return;                  // uniform per wave: EXEC stays all-1
  const int m = lane & 15;
  int row = tile * 16 + m;
  if (row >= V_) row = V_ - 1;                  // clamp loads for partial last tile
  const float* rowp = W + (size_t)row * H_ + ((lane >> 4) << 2);
  const bool lo = (lane < 16);
  v8f acc = {};
#pragma unroll 4
  for (int kb = 0; kb < H_; kb += 8) {
    float4 wa = *(const float4*)(rowp + kb);        // W[row][kb + (lo?0:4) .. +3]
    float4 ha = *(const float4*)(h + kb);           // uniform
    float4 hb = *(const float4*)(h + kb + 4);       // uniform
    v2f a1 = { wa.x, wa.y };
    v2f a2 = { wa.z, wa.w };
    v2f b1 = { lo ? ha.x : ha.y, lo ? hb.x : hb.y };
    v2f b2 = { lo ? ha.z : ha.w, lo ? hb.z : hb.w };
    acc = __builtin_amdgcn_wmma_f32_16x16x4_f32(false, a1, false, b1,
                                                (short)0, acc, false, false);
    acc = __builtin_amdgcn_wmma_f32_16x16x4_f32(false, a2, false, b2,
                                                (short)0, acc, false, false);
  }
  // D layout: VGPR j, lanes 0-15 -> M=j ; lanes 16-31 -> M=8+j (all N equal)
  if ((lane & 15) == 0) {
    int base = tile * 16 + (lane >> 4) * 8;
#pragma unroll
    for (int j = 0; j < 8; ++j) {
      int r = base + j;
      if (r < V_) out[r] = acc[j] + bias[r];
    }
  }
}

// ---------------------------------------------------------------------------
// Kernel 9: log-softmax over V (single block, 1024 threads; data sits in L2)
// ---------------------------------------------------------------------------
__global__ void log_softmax_k(const float* __restrict__ x,
                              float* __restrict__ y, int n) {
  __shared__ float red[1024];
  int tid = threadIdx.x;
  float m = -3.4e38f;
  for (int i = tid; i < n; i += 1024) m = fmaxf(m, x[i]);
  red[tid] = m; __syncthreads();
  for (int s = 512; s > 0; s >>= 1) {
    if (tid < s) red[tid] = fmaxf(red[tid], red[tid + s]);
    __syncthreads();
  }
  float mx = red[0]; __syncthreads();
  float sum = 0.0f;
  for (int i = tid; i < n; i += 1024) sum += expf(x[i] - mx);
  red[tid] = sum; __syncthreads();
  for (int s = 512; s > 0; s >>= 1) {
    if (tid < s) red[tid] += red[tid + s];
    __syncthreads();
  }
  float lg = mx + logf(red[0]);
  for (int i = tid; i < n; i += 1024) y[i] = x[i] - lg;
}

// ---------------------------------------------------------------------------
extern "C" void kernel_launch(void* const* d_in, const int* in_sizes, int n_in,
                              void* d_out, int out_size, void* d_ws,
                              size_t ws_size, hipStream_t stream) {
  const int*   tok    = (const int*)  d_in[0];
  const float* hid    = (const float*)d_in[1];
  const float* enc    = (const float*)d_in[2];
  const float* emb    = (const float*)d_in[3];
  const float* attn_W = (const float*)d_in[4];
  const float* attn_b = (const float*)d_in[5];
  const float* comb_W = (const float*)d_in[6];
  const float* comb_b = (const float*)d_in[7];
  const float* W_ih   = (const float*)d_in[8];
  const float* W_hh   = (const float*)d_in[9];
  const float* b_ih   = (const float*)d_in[10];
  const float* b_hh   = (const float*)d_in[11];
  const float* out_W  = (const float*)d_in[12];
  const float* out_b  = (const float*)d_in[13];

  float* out = (float*)d_out;            // [V] log-probs | [H] h_new | [F] attn_w
  float* ws  = (float*)d_ws;

  // workspace layout (floats, all 16B aligned)
  float* cat1  = ws;               // 2048
  float* cat2  = ws + 2048;        // 2048  ([emb, attn_applied])
  float* alog  = ws + 4096;        // 2048  attention logits
  float* xbuf  = ws + 6144;        // 1024  relu(combine)
  float* gx    = ws + 7168;        // 3072
  float* gh    = ws + 10240;       // 3072
  float* lgts  = ws + 13312;       // 50257 (padded to 50432)
  float* hn_ws = ws + 63744;       // 1024  aligned copy of h_new

  float* aw = out + V_ + H_;       // attn_weights output slot
  float* hn = out + V_;            // h_new output slot

  // 1) embed + concat buffers
  prep_kernel<<<(2 * H_) / 256, 256, 0, stream>>>(tok, emb, hid, cat1, cat2);
  // 2) attention logits: [F x 2H] GEMV
  gemv_f32<<<F_ / 8, 256, 0, stream>>>(attn_W, cat1, attn_b, alog, F_, 2 * H_, 0);
  // 3) softmax -> attn_weights (third output)
  softmax_k<<<1, 256, 0, stream>>>(alog, aw, F_);
  // 4) attn_applied = attn_weights @ encoder_output  (into cat2[H:2H])
  attn_apply<<<dim3(4, 8), 256, 0, stream>>>(enc, aw, cat2 + H_);
  // 5) x = relu(comb_W @ cat2 + comb_b): [H x 2H]
  gemv_f32<<<H_ / 8, 256, 0, stream>>>(comb_W, cat2, comb_b, xbuf, H_, 2 * H_, 1);
  // 6) GRU input/hidden projections: [3H x H] each
  gemv_f32<<<(3 * H_) / 8, 256, 0, stream>>>(W_ih, xbuf, b_ih, gx, 3 * H_, H_, 0);
  gemv_f32<<<(3 * H_) / 8, 256, 0, stream>>>(W_hh, hid,  b_hh, gh, 3 * H_, H_, 0);
  // 7) gates -> h_new (second output + aligned ws copy)
  gru_gates<<<(H_ + 255) / 256, 256, 0, stream>>>(gx, gh, hid, hn, hn_ws);
  // 8) vocab projection via WMMA: [V x H], 16 rows / wave
  int tiles = (V_ + 15) / 16;                       // 3142
  outproj_wmma<<<(tiles + 7) / 8, 256, 0, stream>>>(out_W, hn_ws, out_b, lgts);
  // 9) log-softmax over V -> first output
  log_softmax_k<<<1, 1024, 0, stream>>>(lgts, out, V_);
}